// TransGNN_35708358099446
// MI455X (gfx1250) — compile-verified
//
#include <hip/hip_runtime.h>
#include <math.h>

typedef __attribute__((ext_vector_type(2))) float v2f;
typedef __attribute__((ext_vector_type(8))) float v8f;

#define D128 128
#define HEADS 4
#define DH 32
#define KSAMP 20
#define LN_EPS 1e-5f

__device__ __forceinline__ v8f wmma_f32_4(v2f a, v2f b, v8f c) {
    // D = A(16x4 f32) * B(4x16 f32) + C(16x16 f32)
    return __builtin_amdgcn_wmma_f32_16x16x4_f32(
        /*neg_a=*/false, a, /*neg_b=*/false, b,
        /*c_mod=*/(short)0, c, /*reuse_a=*/false, /*reuse_b=*/false);
}

// ---------------------------------------------------------------------------
// GEMM: out[n, NOUT] = A[n, KD] @ W^T + bias   (W is [NOUT, KD] row-major)
// 256 threads = 8 waves. Each block does 16 rows x NOUT cols. Wave w handles
// NOUT/128 column tiles. A strip staged in LDS (padded stride).
// ---------------------------------------------------------------------------
template <int KD, int NOUT, bool RELU>
__global__ __launch_bounds__(256) void tg_gemm_bias(
    const float* __restrict__ A, const float* __restrict__ W,
    const float* __restrict__ bias, float* __restrict__ out, int n) {
    constexpr int LDA = KD + 4;
    __shared__ float sA[16 * LDA];
    const int row0 = blockIdx.x * 16;
    const int tid = threadIdx.x;

    for (int i = tid; i < 16 * KD; i += 256) {
        int r = i / KD, c = i % KD;
        int gr = row0 + r;
        sA[r * LDA + c] = (gr < n) ? A[(size_t)gr * KD + c] : 0.0f;
    }
    __syncthreads();

    const int wave = tid >> 5;
    const int lane = tid & 31;
    const int m = lane & 15;          // A row / B col within tile
    const int khalf = (lane >> 4) * 2;  // K sub-slot per ISA frag layout
    constexpr int TILES = NOUT / 16 / 8;

    const int nn = lane & 15;
    const int mbase = (lane >> 4) * 8;

    for (int t = 0; t < TILES; ++t) {
        const int col0 = (wave * TILES + t) * 16;
        v8f c = 0.0f;
        const float* __restrict__ wrow = W + (size_t)(col0 + m) * KD;
        for (int k = 0; k < KD; k += 4) {
            v2f a, b;
            a.x = sA[m * LDA + k + khalf];
            a.y = sA[m * LDA + k + khalf + 1];
            b.x = wrow[k + khalf];
            b.y = wrow[k + khalf + 1];
            c = wmma_f32_4(a, b, c);
        }
        float bval = bias[col0 + nn];
        for (int v = 0; v < 8; ++v) {
            int gr = row0 + mbase + v;
            if (gr < n) {
                float val = c[v] + bval;
                if (RELU) val = fmaxf(val, 0.0f);
                out[(size_t)gr * NOUT + col0 + nn] = val;
            }
        }
    }
}

// ---------------------------------------------------------------------------
// GEMM (NOUT=128) with fused epilogue: y = gemm(A)+bias; (opt relu);
// x = y + resid; out = LayerNorm(x)*gamma+beta.
// CONCAT: A = [A1 | A2], each [n, KD/2].
// ---------------------------------------------------------------------------
template <int KD, bool RELU, bool CONCAT>
__global__ __launch_bounds__(256) void tg_gemm_ln(
    const float* __restrict__ A1, const float* __restrict__ A2,
    const float* __restrict__ W, const float* __restrict__ bias,
    const float* __restrict__ resid, const float* __restrict__ gamma,
    const float* __restrict__ beta, float* __restrict__ out, int n) {
    constexpr int LDA = KD + 4;
    __shared__ float sA[16 * LDA];
    __shared__ float sO[16 * 132];
    const int row0 = blockIdx.x * 16;
    const int tid = threadIdx.x;

    for (int i = tid; i < 16 * KD; i += 256) {
        int r = i / KD, c = i % KD;
        int gr = row0 + r;
        float v = 0.0f;
        if (gr < n) {
            if (!CONCAT)
                v = A1[(size_t)gr * KD + c];
            else
                v = (c < KD / 2) ? A1[(size_t)gr * (KD / 2) + c]
                                 : A2[(size_t)gr * (KD / 2) + (c - KD / 2)];
        }
        sA[r * LDA + c] = v;
    }
    __syncthreads();

    const int wave = tid >> 5;
    const int lane = tid & 31;
    const int m = lane & 15;
    const int khalf = (lane >> 4) * 2;
    const int col0 = wave * 16;  // 8 waves * 16 = 128 cols

    v8f c = 0.0f;
    const float* __restrict__ wrow = W + (size_t)(col0 + m) * KD;
    for (int k = 0; k < KD; k += 4) {
        v2f a, b;
        a.x = sA[m * LDA + k + khalf];
        a.y = sA[m * LDA + k + khalf + 1];
        b.x = wrow[k + khalf];
        b.y = wrow[k + khalf + 1];
        c = wmma_f32_4(a, b, c);
    }
    {
        const int nn = lane & 15;
        const int mbase = (lane >> 4) * 8;
        float bval = bias[col0 + nn];
        for (int v = 0; v < 8; ++v) {
            float val = c[v] + bval;
            if (RELU) val = fmaxf(val, 0.0f);
            sO[(mbase + v) * 132 + col0 + nn] = val;
        }
    }
    __syncthreads();

    // LayerNorm: 16 threads per row (row = tid>>4), 8 cols each.
    const int r = tid >> 4;
    const int cbase = (tid & 15) * 8;
    const int gr = row0 + r;
    float vals[8];
    float s = 0.0f, s2 = 0.0f;
    for (int j = 0; j < 8; ++j) {
        float rv = (gr < n) ? resid[(size_t)gr * D128 + cbase + j] : 0.0f;
        float x = sO[r * 132 + cbase + j] + rv;
        vals[j] = x;
        s += x;
        s2 += x * x;
    }
    for (int o = 1; o < 16; o <<= 1) {
        s += __shfl_xor(s, o, 32);
        s2 += __shfl_xor(s2, o, 32);
    }
    float mean = s * (1.0f / 128.0f);
    float var = s2 * (1.0f / 128.0f) - mean * mean;
    float inv = rsqrtf(var + LN_EPS);
    if (gr < n) {
        for (int j = 0; j < 8; ++j) {
            int cc = cbase + j;
            out[(size_t)gr * D128 + cc] =
                (vals[j] - mean) * inv * gamma[cc] + beta[cc];
        }
    }
}

// ---------------------------------------------------------------------------
// Attention: one wave per node, online softmax over KSAMP gathered K/V rows.
// qkv layout per node: [q(128) | k(128) | v(128)]. Lane l holds dim l of each
// of the 4 heads. Gathered rows are L2-resident (projected table is 76.8 MB).
// ---------------------------------------------------------------------------
__global__ __launch_bounds__(256) void tg_attn(
    const float* __restrict__ qkv, const int* __restrict__ samples,
    float* __restrict__ attn, int n) {
    const int node = blockIdx.x * 8 + (threadIdx.x >> 5);
    const int lane = threadIdx.x & 31;
    if (node >= n) return;

    const float* q = qkv + (size_t)node * 384;
    float qv[HEADS], m[HEADS], l[HEADS], acc[HEADS];
    for (int h = 0; h < HEADS; ++h) {
        qv[h] = q[h * DH + lane];
        m[h] = -1.0e30f;
        l[h] = 0.0f;
        acc[h] = 0.0f;
    }
    const float scale = 0.17677669529663688f;  // 1/sqrt(32)

    for (int j = 0; j < KSAMP; ++j) {
        int sidx = samples[(size_t)node * KSAMP + j];
        const float* kr = qkv + (size_t)sidx * 384 + 128;
        const float* vr = qkv + (size_t)sidx * 384 + 256;
        for (int h = 0; h < HEADS; ++h) {
            float p = qv[h] * kr[h * DH + lane];
            for (int o = 1; o < 32; o <<= 1) p += __shfl_xor(p, o, 32);
            p *= scale;
            float vv = vr[h * DH + lane];
            float nm = fmaxf(m[h], p);
            float sc = __expf(m[h] - nm);
            float e = __expf(p - nm);
            acc[h] = acc[h] * sc + e * vv;
            l[h] = l[h] * sc + e;
            m[h] = nm;
        }
    }
    for (int h = 0; h < HEADS; ++h)
        attn[(size_t)node * D128 + h * DH + lane] = acc[h] / l[h];
}

// ---------------------------------------------------------------------------
// GNN scatter: neigh[src[e]] += h[dst[e]]  (one wave per edge, 4 f32/lane)
// ---------------------------------------------------------------------------
__global__ __launch_bounds__(256) void tg_zero(float* __restrict__ p, int count) {
    int i = blockIdx.x * 256 + threadIdx.x;
    int stride = gridDim.x * 256;
    for (; i < count; i += stride) p[i] = 0.0f;
}

__global__ __launch_bounds__(256) void tg_scatter(
    const float* __restrict__ h, const int* __restrict__ src,
    const int* __restrict__ dst, float* __restrict__ neigh, int e) {
    const int edge = blockIdx.x * 8 + (threadIdx.x >> 5);
    const int lane = threadIdx.x & 31;
    if (edge >= e) return;
    const int s = src[edge];
    const int d = dst[edge];
    const float* hr = h + (size_t)d * D128;
    float* nr = neigh + (size_t)s * D128;
    for (int j = 0; j < 4; ++j)
        atomicAdd(nr + lane + j * 32, hr[lane + j * 32]);
}

// ---------------------------------------------------------------------------
extern "C" void kernel_launch(void* const* d_in, const int* in_sizes, int n_in,
                              void* d_out, int out_size, void* d_ws,
                              size_t ws_size, hipStream_t stream) {
    const float* x = (const float*)d_in[0];
    const int* samples = (const int*)d_in[1];
    const int* esrc = (const int*)d_in[2];
    const int* edst = (const int*)d_in[3];
    const float* t_in_w = (const float*)d_in[4];
    const float* t_in_b = (const float*)d_in[5];
    const float* t_out_w = (const float*)d_in[6];
    const float* t_out_b = (const float*)d_in[7];
    const float* t_w1 = (const float*)d_in[8];
    const float* t_b1 = (const float*)d_in[9];
    const float* t_w2 = (const float*)d_in[10];
    const float* t_b2 = (const float*)d_in[11];
    const float* ln1g = (const float*)d_in[12];
    const float* ln1b = (const float*)d_in[13];
    const float* ln2g = (const float*)d_in[14];
    const float* ln2b = (const float*)d_in[15];
    const float* g_w = (const float*)d_in[16];
    const float* g_b = (const float*)d_in[17];
    const float* glng = (const float*)d_in[18];
    const float* glnb = (const float*)d_in[19];

    const int N = in_sizes[0] / D128;
    const int E = in_sizes[2];

    // Workspace layout (floats). region1 (N*512) is shared by QKV (N*384)
    // and FFN-mid (N*512); attnb (N*128) is shared with GNN neigh.
    float* ws = (float*)d_ws;
    float* region1 = ws;                           // N*512
    float* attnb = region1 + (size_t)N * 512;      // N*128 (also neigh)
    float* hA = attnb + (size_t)N * D128;          // N*128
    float* hB = hA + (size_t)N * D128;             // N*128
    float* hC = hB + (size_t)N * D128;             // N*128 (h1 temp)

    const int rb = (N + 15) / 16;        // 16-row strips
    const int wb = (N + 7) / 8;          // 8 waves per block
    const int eb = (E + 7) / 8;

    auto trans = [&](const float* hin, float* hout, int i) {
        // QKV = hin @ in_w^T + in_b  -> region1[N,384]
        tg_gemm_bias<128, 384, false><<<rb, 256, 0, stream>>>(
            hin, t_in_w + (size_t)i * 384 * 128, t_in_b + (size_t)i * 384,
            region1, N);
        // attention over gathered projected K/V (L2-resident)
        tg_attn<<<wb, 256, 0, stream>>>(region1, samples, attnb, N);
        // h1 = LN(hin + attn @ out_w^T + out_b)
        tg_gemm_ln<128, false, false><<<rb, 256, 0, stream>>>(
            attnb, nullptr, t_out_w + (size_t)i * 128 * 128,
            t_out_b + (size_t)i * 128, hin, ln1g + (size_t)i * 128,
            ln1b + (size_t)i * 128, hC, N);
        // mid = relu(h1 @ w1^T + b1) -> region1[N,512]  (QKV dead)
        tg_gemm_bias<128, 512, true><<<rb, 256, 0, stream>>>(
            hC, t_w1 + (size_t)i * 512 * 128, t_b1 + (size_t)i * 512, region1,
            N);
        // hout = LN(h1 + mid @ w2^T + b2)
        tg_gemm_ln<512, false, false><<<rb, 256, 0, stream>>>(
            region1, nullptr, t_w2 + (size_t)i * 128 * 512,
            t_b2 + (size_t)i * 128, hC, ln2g + (size_t)i * 128,
            ln2b + (size_t)i * 128, hout, N);
    };

    auto gnn = [&](const float* hin, float* hout, int i) {
        tg_zero<<<2048, 256, 0, stream>>>(attnb, N * D128);
        tg_scatter<<<eb, 256, 0, stream>>>(hin, esrc, edst, attnb, E);
        // hout = LN(relu([hin|neigh] @ g_w^T + g_b) + hin)
        tg_gemm_ln<256, true, true><<<rb, 256, 0, stream>>>(
            hin, attnb, g_w + (size_t)i * 128 * 256, g_b + (size_t)i * 128,
            hin, glng + (size_t)i * 128, glnb + (size_t)i * 128, hout, N);
    };

    trans(x, hA, 0);
    gnn(hA, hB, 0);
    trans(hB, hA, 1);
    gnn(hA, hB, 1);
    trans(hB, (float*)d_out, 2);
}